// Top1Gate_64433099374669
// MI455X (gfx1250) — compile-verified
//
#include <hip/hip_runtime.h>

typedef __attribute__((ext_vector_type(2))) float v2f;
typedef __attribute__((ext_vector_type(4))) float v4f;
typedef __attribute__((ext_vector_type(8))) float v8f;

#define NTOK 8192
#define DIM  2048
#define NEXP 64
#define CAP  128
#define SEC  (8192LL * 64LL * 128LL)   // 67,108,864 elements per [S,E,C] tensor

// ---------------- kernel 0: zero the small accumulators ----------------
__global__ void init_acc_kernel(float* __restrict__ me_sum, unsigned* __restrict__ ce_cnt) {
  const int t = threadIdx.x;
  if (t < NEXP) { me_sum[t] = 0.0f; ce_cnt[t] = 0u; }
}

// ---------------- kernel 1: WMMA gating (logits -> softmax -> argmax) ----------------
// Block = 256 threads = 8 waves. Each wave computes a 16-token x 64-expert logits
// tile with V_WMMA_F32_16X16X4_F32 (4 N-tiles, K stepped by 4 over D=2048).
__launch_bounds__(256)
__global__ void gate_wmma_kernel(const float* __restrict__ x, const float* __restrict__ wg,
                                 float* __restrict__ me_sum, unsigned* __restrict__ ce_cnt,
                                 int* __restrict__ expert_s, float* __restrict__ gate_s)
{
  __shared__ float lds_logits[8][16][NEXP];   // 32 KB
  __shared__ float lds_max[8][16];
  __shared__ float lds_sum[8][16];

  const int wave = threadIdx.x >> 5;
  const int lane = threadIdx.x & 31;
  const int half = lane >> 4;     // 0: K = k,k+1 ; 1: K = k+2,k+3
  const int l15  = lane & 15;
  const int m0   = blockIdx.x * 128 + wave * 16;

  // A fragment: 16x4 f32, lane row = token, 2 consecutive K per lane half
  const float* xrow = x + (size_t)(m0 + l15) * DIM + (half << 1);
  // B fragments: B[k][n] = wg[n][k]; lane column = expert within 16-wide tile
  const float* w0 = wg + (size_t)l15 * DIM + (half << 1);
  const float* w1 = w0 + (size_t)16 * DIM;
  const float* w2 = w1 + (size_t)16 * DIM;
  const float* w3 = w2 + (size_t)16 * DIM;

  v8f c0 = {}, c1 = {}, c2 = {}, c3 = {};

  #pragma unroll 4
  for (int k = 0; k < DIM; k += 4) {
    const v2f a  = *(const v2f*)(xrow + k);
    const v2f b0 = *(const v2f*)(w0 + k);
    const v2f b1 = *(const v2f*)(w1 + k);
    const v2f b2 = *(const v2f*)(w2 + k);
    const v2f b3 = *(const v2f*)(w3 + k);
    c0 = __builtin_amdgcn_wmma_f32_16x16x4_f32(false, a, false, b0, (short)0, c0, false, false);
    c1 = __builtin_amdgcn_wmma_f32_16x16x4_f32(false, a, false, b1, (short)0, c1, false, false);
    c2 = __builtin_amdgcn_wmma_f32_16x16x4_f32(false, a, false, b2, (short)0, c2, false, false);
    c3 = __builtin_amdgcn_wmma_f32_16x16x4_f32(false, a, false, b3, (short)0, c3, false, false);
  }

  // C/D layout: VGPR r -> M = (lane<16 ? r : 8+r), N = lane&15
  const int mb = half << 3;
  #pragma unroll
  for (int r = 0; r < 8; ++r) {
    lds_logits[wave][mb + r][l15 +  0] = c0[r];
    lds_logits[wave][mb + r][l15 + 16] = c1[r];
    lds_logits[wave][mb + r][l15 + 32] = c2[r];
    lds_logits[wave][mb + r][l15 + 48] = c3[r];
  }
  __syncthreads();

  // per-token softmax stats: max, argmax (first max, like jnp.argmax), exp-sum
  if (lane < 16) {
    const float* row = lds_logits[wave][lane];
    float mx = row[0]; int am = 0;
    #pragma unroll
    for (int e = 1; e < NEXP; ++e) { const float v = row[e]; if (v > mx) { mx = v; am = e; } }
    float sum = 0.0f;
    #pragma unroll
    for (int e = 0; e < NEXP; ++e) sum += __expf(row[e] - mx);
    const int s = m0 + lane;
    expert_s[s] = am;
    gate_s[s]   = 1.0f / sum;          // softmax value at the argmax
    atomicAdd(&ce_cnt[am], 1u);
    lds_max[wave][lane] = mx;
    lds_sum[wave][lane] = sum;
  }
  __syncthreads();

  // me[e] partials: each lane owns experts (lane, lane+32) over this wave's 16 tokens
  float p0 = 0.0f, p1 = 0.0f;
  #pragma unroll
  for (int tok = 0; tok < 16; ++tok) {
    const float mx  = lds_max[wave][tok];
    const float inv = 1.0f / lds_sum[wave][tok];
    p0 += __expf(lds_logits[wave][tok][lane]      - mx) * inv;
    p1 += __expf(lds_logits[wave][tok][lane + 32] - mx) * inv;
  }
  atomicAdd(&me_sum[lane],      p0);
  atomicAdd(&me_sum[lane + 32], p1);
}

// ---------------- kernel 2: order-preserving per-expert prefix (cumsum - 1) ----------------
// Single 1024-thread workgroup; stable within-chunk ranking via per-expert bitmasks.
__launch_bounds__(1024)
__global__ void scan_loc_kernel(const int* __restrict__ expert_s, int* __restrict__ loc_s)
{
  __shared__ unsigned bits[NEXP * 32];   // 64 experts x 1024-bit masks = 8 KB
  __shared__ int base[NEXP];
  const int t = threadIdx.x;
  if (t < NEXP) base[t] = 0;

  for (int chunk = 0; chunk < NTOK; chunk += 1024) {
    bits[t] = 0u; bits[t + 1024] = 0u;
    __syncthreads();

    const int e = expert_s[chunk + t];
    atomicOr(&bits[e * 32 + (t >> 5)], 1u << (t & 31));
    __syncthreads();

    // stable rank = number of earlier tokens (lower thread id) with same expert
    int rank = 0;
    const int w = t >> 5;
    for (int i = 0; i < w; ++i) rank += __popc(bits[e * 32 + i]);
    rank += __popc(bits[e * 32 + w] & ((1u << (t & 31)) - 1u));

    const int loc = base[e] + rank;
    loc_s[chunk + t] = (loc < CAP) ? loc : -1;   // capacity drop
    __syncthreads();

    if (t < NEXP) {
      int tot = 0;
      #pragma unroll
      for (int i = 0; i < 32; ++i) tot += __popc(bits[t * 32 + i]);
      base[t] += tot;
    }
    __syncthreads();
  }
}

// ---------------- kernel 3: NT zero-fill of the 512 MB output ----------------
__launch_bounds__(256)
__global__ void zero_out_kernel(float* __restrict__ out, const long long n)
{
  const long long stride = (long long)gridDim.x * blockDim.x;
  const long long i = (long long)blockIdx.x * blockDim.x + threadIdx.x;
  const long long n4 = n >> 2;
  const v4f z = {0.0f, 0.0f, 0.0f, 0.0f};
  v4f* out4 = (v4f*)out;
  for (long long j = i; j < n4; j += stride)
    __builtin_nontemporal_store(z, out4 + j);   // stream past L2 (output >> 192MB L2)
  if (i == 0)
    for (long long j = n4 << 2; j < n; ++j) out[j] = 0.0f;
}

// ---------------- kernel 4: aux loss ----------------
__global__ void laux_kernel(const float* __restrict__ me_sum, const unsigned* __restrict__ ce_cnt,
                            float* __restrict__ out)
{
  __shared__ float red[NEXP];
  const int t = threadIdx.x;
  red[t] = me_sum[t] * (float)ce_cnt[t];
  __syncthreads();
  for (int s = NEXP / 2; s > 0; s >>= 1) {
    if (t < s) red[t] += red[t + s];
    __syncthreads();
  }
  // l_aux = mean_e(me*ce)*E^2 = (E / S^2) * sum_e me_sum[e]*ce_cnt[e]
  if (t == 0) out[0] = red[0] * ((float)NEXP / ((float)NTOK * (float)NTOK));
}

// ---------------- kernel 5: sparse scatter of combine + dispatch ----------------
__global__ void scatter_kernel(const int* __restrict__ expert_s, const float* __restrict__ gate_s,
                               const int* __restrict__ loc_s, float* __restrict__ out)
{
  const int s = blockIdx.x * blockDim.x + threadIdx.x;
  if (s >= NTOK) return;
  const int loc = loc_s[s];
  if (loc < 0) return;
  const long long idx = 1LL + (long long)s * (NEXP * CAP) + (long long)expert_s[s] * CAP + loc;
  out[idx]       = gate_s[s];   // combine1_sec
  out[idx + SEC] = 1.0f;        // dispatch_mask (bool -> 1.0f)
}

extern "C" void kernel_launch(void* const* d_in, const int* in_sizes, int n_in,
                              void* d_out, int out_size, void* d_ws, size_t ws_size,
                              hipStream_t stream) {
  (void)in_sizes; (void)n_in; (void)out_size; (void)ws_size;
  const float* x  = (const float*)d_in[0];   // [8192, 2048] f32
  const float* wg = (const float*)d_in[1];   // [64, 2048]   f32
  float* out = (float*)d_out;                // [1 + 2*S*E*C] f32

  char* ws = (char*)d_ws;
  float*    me_sum   = (float*)(ws);
  unsigned* ce_cnt   = (unsigned*)(ws + 256);
  int*      expert_s = (int*)(ws + 512);
  float*    gate_s   = (float*)(ws + 512 + 4 * NTOK);
  int*      loc_s    = (int*)(ws + 512 + 8 * NTOK);

  init_acc_kernel<<<1, 64, 0, stream>>>(me_sum, ce_cnt);
  gate_wmma_kernel<<<NTOK / 128, 256, 0, stream>>>(x, wg, me_sum, ce_cnt, expert_s, gate_s);
  scan_loc_kernel<<<1, 1024, 0, stream>>>(expert_s, loc_s);
  const long long n_out = 1LL + 2LL * SEC;
  zero_out_kernel<<<4096, 256, 0, stream>>>(out, n_out);
  laux_kernel<<<1, NEXP, 0, stream>>>(me_sum, ce_cnt, out);
  scatter_kernel<<<(NTOK + 255) / 256, 256, 0, stream>>>(expert_s, gate_s, loc_s, out);
}